// A2C_26938034880704
// MI455X (gfx1250) — compile-verified
//
#include <hip/hip_runtime.h>
#include <hip/hip_bf16.h>
#include <math.h>

typedef __attribute__((ext_vector_type(2))) float v2f;
typedef __attribute__((ext_vector_type(8))) float v8f;

#define CCH 21   // in_channels
#define PAD 24   // padded row stride (16B-aligned rows)
#define HID 32   // MLP hidden

// ---------------------------------------------------------------------------
// K0: deg = 1 (self-loop), zero critic sum accumulator
// ---------------------------------------------------------------------------
__global__ void k_init(float* __restrict__ deg, float* __restrict__ csum, int n) {
  int i = blockIdx.x * blockDim.x + threadIdx.x;
  if (i < n)  deg[i]  = 1.0f;
  if (i < 32) csum[i] = 0.0f;
}

// ---------------------------------------------------------------------------
// K1: in-degree over dst (float atomics; exact for counts < 2^24)
// ---------------------------------------------------------------------------
__global__ void k_deg(const int* __restrict__ dst, float* __restrict__ deg, int e) {
  int i = blockIdx.x * blockDim.x + threadIdx.x;
  if (i < e) unsafeAtomicAdd(&deg[dst[i]], 1.0f);
}

// ---------------------------------------------------------------------------
// K2: dinv = rsqrt(deg); hs = dinv * (x @ W.T) for both convs;
//     seed accumulators with the self-loop term (acc = hs).
// ---------------------------------------------------------------------------
__global__ void k_conv(const float* __restrict__ x, const float* __restrict__ deg,
                       const float* __restrict__ Wa, const float* __restrict__ Wc,
                       float* __restrict__ dinv,
                       float* __restrict__ hsa, float* __restrict__ hsc,
                       float* __restrict__ acca, float* __restrict__ accc, int n) {
  int i = blockIdx.x * blockDim.x + threadIdx.x;
  if (i >= n) return;
  float xi[CCH];
  #pragma unroll
  for (int k = 0; k < CCH; ++k) xi[k] = x[(size_t)i * CCH + k];
  float di = rsqrtf(deg[i]);
  dinv[i] = di;
  size_t ro = (size_t)i * PAD;
  for (int o = 0; o < CCH; ++o) {
    float sa = 0.f, sc = 0.f;
    #pragma unroll
    for (int k = 0; k < CCH; ++k) {
      sa = fmaf(xi[k], Wa[o * CCH + k], sa);
      sc = fmaf(xi[k], Wc[o * CCH + k], sc);
    }
    sa *= di; sc *= di;
    hsa[ro + o] = sa; acca[ro + o] = sa;
    hsc[ro + o] = sc; accc[ro + o] = sc;
  }
}

// ---------------------------------------------------------------------------
// K3 (dominant): fused edge scatter for both convs. Working set (~38 MB)
// is L2-resident on MI455X (192 MB), so atomics resolve at L2 bandwidth.
// ---------------------------------------------------------------------------
__global__ void k_edge(const int* __restrict__ src, const int* __restrict__ dst,
                       const float* __restrict__ hsa, const float* __restrict__ hsc,
                       float* __restrict__ acca, float* __restrict__ accc, int e) {
  int i = blockIdx.x * blockDim.x + threadIdx.x;
  if (i >= e) return;
  int s = src[i], d = dst[i];
  const float4* ra = (const float4*)(hsa + (size_t)s * PAD);  // 16B-aligned rows
  const float4* rc = (const float4*)(hsc + (size_t)s * PAD);
  float* wa = acca + (size_t)d * PAD;
  float* wc = accc + (size_t)d * PAD;
  #pragma unroll
  for (int q = 0; q < 5; ++q) {                 // channels 0..19 as b128 loads
    float4 va = ra[q], vc = rc[q];
    unsafeAtomicAdd(&wa[q * 4 + 0], va.x);
    unsafeAtomicAdd(&wa[q * 4 + 1], va.y);
    unsafeAtomicAdd(&wa[q * 4 + 2], va.z);
    unsafeAtomicAdd(&wa[q * 4 + 3], va.w);
    unsafeAtomicAdd(&wc[q * 4 + 0], vc.x);
    unsafeAtomicAdd(&wc[q * 4 + 1], vc.y);
    unsafeAtomicAdd(&wc[q * 4 + 2], vc.z);
    unsafeAtomicAdd(&wc[q * 4 + 3], vc.w);
  }
  unsafeAtomicAdd(&wa[20], ((const float*)ra)[20]);  // channel 20
  unsafeAtomicAdd(&wc[20], ((const float*)rc)[20]);
}

// ---------------------------------------------------------------------------
// K4: per-node epilogue. a = relu(dinv*accA + ba) + x (actor features),
//     c = relu(dinv*accC + bc) + x (critic features, block-reduced to csum),
//     actor MLP head via full-precision V_WMMA_F32_16X16X4_F32 (16 nodes/wave,
//     K-blocked accumulation through the C operand), softplus -> concentration.
// ---------------------------------------------------------------------------
__global__ __launch_bounds__(256) void k_post(
    const float* __restrict__ x, const float* __restrict__ dinv,
    const float* __restrict__ acca, const float* __restrict__ accc,
    const float* __restrict__ ba, const float* __restrict__ bc,
    const float* __restrict__ w1, const float* __restrict__ b1,
    const float* __restrict__ w2, const float* __restrict__ b2,
    const float* __restrict__ w3, const float* __restrict__ b3,
    float* __restrict__ conc, float* __restrict__ csum, int n) {
  __shared__ float sX[8][16][HID];   // per-wave input tile (f32); later layer-2 out
  __shared__ float sH[8][16][HID];   // per-wave layer-1 out (f32)
  __shared__ float sCS[32];          // block-local critic channel sums

  const int tid  = threadIdx.x;
  const int wave = tid >> 5;
  const int lane = tid & 31;
  const int base = blockIdx.x * 128 + wave * 16;   // first node of this wave

  if (tid < 32) sCS[tid] = 0.0f;
  // zero tile: covers K-padding (21..31) and out-of-range tail nodes,
  // so EXEC can stay all-ones for the WMMAs.
  #pragma unroll
  for (int i = lane; i < 16 * HID; i += 32)
    sX[wave][i >> 5][i & 31] = 0.0f;
  __syncthreads();

  for (int i = lane; i < 16 * CCH; i += 32) {
    int m  = i / CCH;
    int cc = i - m * CCH;
    int node = base + m;
    if (node < n) {
      float di = dinv[node];
      float xv = x[(size_t)node * CCH + cc];
      float av = fmaxf(fmaf(di, acca[(size_t)node * PAD + cc], ba[cc]), 0.f) + xv;
      float cv = fmaxf(fmaf(di, accc[(size_t)node * PAD + cc], bc[cc]), 0.f) + xv;
      sX[wave][m][cc] = av;
      atomicAdd(&sCS[cc], cv);          // ds_add_f32
    }
  }
  __syncthreads();
  if (tid < CCH) unsafeAtomicAdd(&csum[tid], sCS[tid]);

  const int half = lane >> 4;           // which 16-lane half (K sub-block)
  const int rc16 = lane & 15;           // row (A) / column (B,C)

  // ---- layer 1: [16 x 21(pad24)] @ w1.T -> [16 x 32], two 16-wide N tiles,
  //      6 K-blocks of 4, f32 WMMA with chained C accumulation
  v8f c0 = {}, c1 = {};
  #pragma unroll
  for (int kb = 0; kb < 6; ++kb) {
    v2f A, B0, B1;
    #pragma unroll
    for (int i = 0; i < 2; ++i) {
      int k = kb * 4 + i + 2 * half;    // 32-bit A/B layout: VGPR i, half-of-wave
      A[i]  = sX[wave][rc16][k];
      B0[i] = (k < CCH) ? w1[rc16 * CCH + k]        : 0.f;
      B1[i] = (k < CCH) ? w1[(rc16 + 16) * CCH + k] : 0.f;
    }
    c0 = __builtin_amdgcn_wmma_f32_16x16x4_f32(false, A, false, B0, (short)0, c0, false, false);
    c1 = __builtin_amdgcn_wmma_f32_16x16x4_f32(false, A, false, B1, (short)0, c1, false, false);
  }
  float bb0 = b1[rc16], bb1 = b1[rc16 + 16];
  #pragma unroll
  for (int r = 0; r < 8; ++r) {
    int m = r + 8 * half;               // C/D layout: VGPR r, half
    sH[wave][m][rc16]      = fmaxf(c0[r] + bb0, 0.f);
    sH[wave][m][rc16 + 16] = fmaxf(c1[r] + bb1, 0.f);
  }
  __syncthreads();

  // ---- layer 2: [16 x 32] @ w2.T -> [16 x 32], 8 K-blocks of 4
  v8f d0 = {}, d1 = {};
  #pragma unroll
  for (int kb = 0; kb < 8; ++kb) {
    v2f A, B0, B1;
    #pragma unroll
    for (int i = 0; i < 2; ++i) {
      int k = kb * 4 + i + 2 * half;
      A[i]  = sH[wave][rc16][k];
      B0[i] = w2[rc16 * HID + k];
      B1[i] = w2[(rc16 + 16) * HID + k];
    }
    d0 = __builtin_amdgcn_wmma_f32_16x16x4_f32(false, A, false, B0, (short)0, d0, false, false);
    d1 = __builtin_amdgcn_wmma_f32_16x16x4_f32(false, A, false, B1, (short)0, d1, false, false);
  }
  float cb0 = b2[rc16], cb1 = b2[rc16 + 16];
  #pragma unroll
  for (int r = 0; r < 8; ++r) {
    int m = r + 8 * half;
    sX[wave][m][rc16]      = fmaxf(d0[r] + cb0, 0.f);   // reuse input tile
    sX[wave][m][rc16 + 16] = fmaxf(d1[r] + cb1, 0.f);
  }
  __syncthreads();

  // ---- layer 3 (1 output) + softplus
  if (lane < 16) {
    int node = base + lane;
    if (node < n) {
      float acc = b3[0];
      #pragma unroll
      for (int k = 0; k < HID; ++k)
        acc = fmaf(sX[wave][lane][k], w3[k], acc);
      float sp = (acc > 20.f) ? acc : log1pf(expf(acc));
      conc[node] = sp + 1e-20f;
    }
  }
}

// ---------------------------------------------------------------------------
// K5: critic value head on the [21] channel sum (tiny, one wave)
// ---------------------------------------------------------------------------
__global__ void k_value(const float* __restrict__ csum,
                        const float* __restrict__ w1, const float* __restrict__ b1,
                        const float* __restrict__ w2, const float* __restrict__ b2,
                        const float* __restrict__ w3, const float* __restrict__ b3,
                        float* __restrict__ out) {
  __shared__ float h1[HID], h2[HID];
  int t = threadIdx.x;
  float s = b1[t];
  #pragma unroll
  for (int k = 0; k < CCH; ++k) s = fmaf(csum[k], w1[t * CCH + k], s);
  h1[t] = fmaxf(s, 0.f);
  __syncthreads();
  s = b2[t];
  #pragma unroll
  for (int k = 0; k < HID; ++k) s = fmaf(h1[k], w2[t * HID + k], s);
  h2[t] = fmaxf(s, 0.f);
  __syncthreads();
  if (t == 0) {
    float v = b3[0];
    #pragma unroll
    for (int k = 0; k < HID; ++k) v = fmaf(h2[k], w3[k], v);
    out[0] = v;
  }
}

// ---------------------------------------------------------------------------
extern "C" void kernel_launch(void* const* d_in, const int* in_sizes, int n_in,
                              void* d_out, int out_size, void* d_ws, size_t ws_size,
                              hipStream_t stream) {
  const float* x     = (const float*)d_in[0];
  const int*   ei    = (const int*)d_in[1];
  const float* aW    = (const float*)d_in[2];
  const float* ab    = (const float*)d_in[3];
  const float* a_w1  = (const float*)d_in[4];
  const float* a_b1  = (const float*)d_in[5];
  const float* a_w2  = (const float*)d_in[6];
  const float* a_b2  = (const float*)d_in[7];
  const float* a_w3  = (const float*)d_in[8];
  const float* a_b3  = (const float*)d_in[9];
  const float* cW    = (const float*)d_in[10];
  const float* cb    = (const float*)d_in[11];
  const float* c_w1  = (const float*)d_in[12];
  const float* c_b1  = (const float*)d_in[13];
  const float* c_w2  = (const float*)d_in[14];
  const float* c_b2  = (const float*)d_in[15];
  const float* c_w3  = (const float*)d_in[16];
  const float* c_b3  = (const float*)d_in[17];

  const int n = in_sizes[0] / CCH;    // 100000
  const int e = in_sizes[1] / 2;      // 6400000
  const int* src = ei;
  const int* dst = ei + e;

  float* ws   = (float*)d_ws;
  float* deg  = ws;                           // n
  float* dinv = deg  + n;                     // n
  float* hsa  = dinv + n;                     // n*PAD
  float* hsc  = hsa  + (size_t)n * PAD;
  float* acca = hsc  + (size_t)n * PAD;
  float* accc = acca + (size_t)n * PAD;
  float* csum = accc + (size_t)n * PAD;       // 32

  float* conc = (float*)d_out;                // [n]
  float* val  = conc + n;                     // [1]

  const int B = 256;
  k_init<<<(n + B - 1) / B, B, 0, stream>>>(deg, csum, n);
  k_deg <<<(e + B - 1) / B, B, 0, stream>>>(dst, deg, e);
  k_conv<<<(n + B - 1) / B, B, 0, stream>>>(x, deg, aW, cW, dinv,
                                            hsa, hsc, acca, accc, n);
  k_edge<<<(e + B - 1) / B, B, 0, stream>>>(src, dst, hsa, hsc, acca, accc, e);
  k_post<<<(n + 127) / 128, 256, 0, stream>>>(x, dinv, acca, accc, ab, cb,
                                              a_w1, a_b1, a_w2, a_b2, a_w3, a_b3,
                                              conc, csum, n);
  k_value<<<1, HID, 0, stream>>>(csum, c_w1, c_b1, c_w2, c_b2, c_w3, c_b3, val);
}